// SingularityDetector_78795470012874
// MI455X (gfx1250) — compile-verified
//
#include <hip/hip_runtime.h>
#include <stdint.h>

// ---------------- tiling configuration ----------------
#define TILE_MATS        256                    // matrices per TDM tile (1 per thread)
#define MAT_DW           16                     // dwords per 4x4 fp32 matrix
#define TILE_DW          (TILE_MATS * MAT_DW)   // 4096 dwords = 16 KB per tile
#define MAT_DW_PAD       17                     // TDM pads 1 dword every 16 -> stride 17
#define BUF_DW           (TILE_MATS * MAT_DW_PAD) // 4352 dwords per LDS buffer
#define BUF_BYTES        (BUF_DW * 4)           // 17408 B; x2 buffers = 34816 B LDS
#define TILES_PER_BLOCK  16
#define THREADS          256                    // 8 wave32 waves

typedef uint32_t u32x4 __attribute__((ext_vector_type(4)));
typedef uint32_t u32x8 __attribute__((ext_vector_type(8)));
typedef __attribute__((address_space(3))) float lds_float;

// Issue one TDM tile load: 1-D tensor of `tdim0` dwords starting at the tile's
// global address, tile_dim0 = TILE_DW, with LDS padding of 1 dword every 16
// dwords (matrix m lands at lds dword m*17 -> conflict-free stride-17 reads).
// OOB dwords (past tensor_dim0) read as zero, so the tail tile is safe.
__device__ __forceinline__ void tdm_issue_tile(const float* in, uint32_t lds_byte_addr,
                                               uint64_t tile_dw_base, uint32_t total_dw)
{
    uint64_t gaddr     = (uint64_t)(uintptr_t)in + tile_dw_base * 4ull;
    uint32_t remaining = total_dw - (uint32_t)tile_dw_base;
    uint32_t tdim0     = remaining < (uint32_t)TILE_DW ? remaining : (uint32_t)TILE_DW;

    // D# group 0: count=1, lds_addr, global_addr[56:0], type=2 ("image")
    u32x4 g0 = {0u, 0u, 0u, 0u};
    g0[0] = 0x1u;                                                   // count=1, user desc
    g0[1] = lds_byte_addr;                                          // bits[63:32]
    g0[2] = (uint32_t)gaddr;                                        // bits[95:64]
    g0[3] = ((uint32_t)(gaddr >> 32) & 0x01FFFFFFu) | 0x80000000u;  // addr hi | type=2

    // D# group 1: data_size=4B, pad_enable, pad_interval=16dw(code 3),
    // pad_amount=1dw(code 0), tensor_dim0=tdim0, tensor_dim1=1, tile_dim0=TILE_DW,
    // tile_dim1=1, workgroup_mask=0 (not in a cluster)
    u32x8 g1 = {0u, 0u, 0u, 0u, 0u, 0u, 0u, 0u};
    g1[0] = (2u << 16) | (1u << 20) | (3u << 22);
    g1[1] = (tdim0 & 0xFFFFu) << 16;        // tensor_dim0[15:0]  (bits 63:48)
    g1[2] = (tdim0 >> 16) | (1u << 16);     // tensor_dim0[31:16] | tensor_dim1 lo = 1
    g1[3] = ((uint32_t)TILE_DW) << 16;      // tile_dim0 (bits 127:112)
    g1[4] = 1u;                             // tile_dim1 = 1, tile_dim2 = 0
    g1[5] = tdim0;                          // tensor_dim0_stride lo (unused, benign)

    u32x4 g2 = {1u, 0u, 0u, 0u};            // tensor_dim2 = 1, higher dims unused
    u32x4 g3 = {0u, 0u, 0u, 0u};

    asm volatile("tensor_load_to_lds %0, %1, %2, %3"
                 :: "s"(g0), "s"(g1), "s"(g2), "s"(g3)
                 : "memory");
}

// det + lambda_min(|.|) threshold for one symmetric 4x4 held in LDS (stride-17 slot).
__device__ __forceinline__ float detect_one(const float* a)
{
    // unique entries of the symmetric matrix
    float a00 = a[0],  a01 = a[1],  a02 = a[2],  a03 = a[3];
    float a11 = a[5],  a12 = a[6],  a13 = a[7];
    float a22 = a[10], a23 = a[11];
    float a33 = a[15];

    // 2x2 minors (shared between det and char-poly coefficients)
    float s0 = a00*a11 - a01*a01;
    float s1 = a00*a12 - a02*a01;
    float s2 = a00*a13 - a03*a01;
    float s3 = a01*a12 - a02*a11;
    float s4 = a01*a13 - a03*a11;
    float s5 = a02*a13 - a03*a12;
    float c5 = a22*a33 - a23*a23;
    float c4 = a12*a33 - a23*a13;
    float c3 = a12*a23 - a22*a13;
    float c2 = a02*a33 - a23*a03;
    float c1 = a02*a23 - a22*a03;
    float c0 = a02*a13 - a12*a03;

    float det = s0*c5 - s1*c4 + s2*c3 + s3*c2 - s4*c1 + s5*c0;

    // characteristic polynomial P(x) = x^4 - e1 x^3 + e2 x^2 - e3 x + e4
    float t1 = a11*a33 - a13*a13;
    float t2 = a01*a33 - a13*a03;
    float t3 = a01*a13 - a11*a03;
    float u1 = a11*a22 - a12*a12;
    float u2 = a01*a22 - a12*a02;
    float u3 = a01*a12 - a11*a02;

    float e1 = a00 + a11 + a22 + a33;
    float e2 = s0 + c5 + (a00*a22 - a02*a02) + (a00*a33 - a03*a03) + u1 + t1;
    float e3 = (a11*c5 - a12*c4 + a13*c3)    // principal minor {1,2,3}
             + (a00*c5 - a02*c2 + a03*c1)    // {0,2,3}
             + (a00*t1 - a01*t2 + a03*t3)    // {0,1,3}
             + (a00*u1 - a01*u2 + a02*u3);   // {0,1,2}
    float e4 = det;

    // Gershgorin lower bound on lambda_min
    float p01 = fabsf(a01), p02 = fabsf(a02), p03 = fabsf(a03);
    float p12 = fabsf(a12), p13 = fabsf(a13), p23 = fabsf(a23);
    float L = fminf(fminf(a00 - (p01 + p02 + p03), a11 - (p01 + p12 + p13)),
                    fminf(a22 - (p02 + p12 + p23), a33 - (p03 + p13 + p23)));

    // Safeguarded Newton from the left: for x < lambda_min, P>0 and P'<0 strictly.
    // Step uses hardware v_rcp_f32 (~1 ulp): ample accuracy, since Newton
    // self-corrects and the final compare is against a 1e-6 threshold.
    float c3p = -e1, c2p = e2, c1p = -e3, c0p = e4;
    float k2 = -3.0f * e1, k1 = 2.0f * e2, k0 = -e3;
    float x = L;
#pragma unroll
    for (int it = 0; it < 8; ++it) {
        float pv = fmaf(fmaf(fmaf(x + c3p, x, c2p), x, c1p), x, c0p);
        float dv = fmaf(fmaf(fmaf(4.0f, x, k2), x, k1), x, k0);
        float step = pv * __builtin_amdgcn_rcpf(fminf(dv, -1e-30f)); // dv <= 0 here
        x = (pv > 0.0f) ? (x - step) : x;       // never overshoot / never NaN
    }

    float meas = fminf(fabsf(det), fabsf(x));
    return (meas < 1e-6f) ? 1.0f : 0.0f;
}

__global__ __launch_bounds__(THREADS)
void SingularityDetector_78795470012874_kernel(const float* __restrict__ in,
                                               float* __restrict__ out,
                                               int nmat, int ntiles)
{
    __shared__ float smem[2 * BUF_DW];

    // LDS byte offset of smem for the TDM descriptor (addrspace(3) cast)
    uint32_t lds_base;
    {
        lds_float* lp = (lds_float*)&smem[0];
        lds_base = (uint32_t)(uintptr_t)lp;
    }

    const uint32_t total_dw = (uint32_t)nmat * 16u;
    const int tid = (int)threadIdx.x;
    const bool w0 = (threadIdx.x < 32u);         // wave 0 drives the TDM
    const int tile0 = (int)blockIdx.x * TILES_PER_BLOCK;

    if (w0) {
        tdm_issue_tile(in, lds_base, (uint64_t)tile0 * TILE_DW, total_dw);
    }

    for (int i = 0; i < TILES_PER_BLOCK; ++i) {
        int tile = tile0 + i;
        if (tile >= ntiles) break;               // uniform across the block

        if (w0) {
            int nxt = tile + 1;
            if ((i + 1 < TILES_PER_BLOCK) && (nxt < ntiles)) {
                tdm_issue_tile(in, lds_base + (uint32_t)((i + 1) & 1) * BUF_BYTES,
                               (uint64_t)nxt * TILE_DW, total_dw);
                __builtin_amdgcn_s_wait_tensorcnt(1);  // tile i complete (in-order)
            } else {
                __builtin_amdgcn_s_wait_tensorcnt(0);
            }
        }
        __syncthreads();                          // tile i visible to all 8 waves

        const float* mp = &smem[(i & 1) * BUF_DW + tid * MAT_DW_PAD];
        float flag = detect_one(mp);
        int g = tile * TILE_MATS + tid;
        if (g < nmat) {
            // non-temporal: don't let the 16 MB result stream evict the
            // TDM-fed input stream from L2
            __builtin_nontemporal_store(flag, &out[g]);
        }

        __syncthreads();                          // buffer free before next overwrite
    }
}

extern "C" void kernel_launch(void* const* d_in, const int* in_sizes, int n_in,
                              void* d_out, int out_size, void* d_ws, size_t ws_size,
                              hipStream_t stream)
{
    (void)n_in; (void)out_size; (void)d_ws; (void)ws_size;
    const float* in = (const float*)d_in[0];
    float* out = (float*)d_out;

    int nmat    = in_sizes[0] / 16;                               // 4,194,304
    int ntiles  = (nmat + TILE_MATS - 1) / TILE_MATS;             // 16384
    int nblocks = (ntiles + TILES_PER_BLOCK - 1) / TILES_PER_BLOCK; // 1024

    SingularityDetector_78795470012874_kernel<<<nblocks, THREADS, 0, stream>>>(
        in, out, nmat, ntiles);
}